// PairwisePolicy_53635551592587
// MI455X (gfx1250) — compile-verified
//
#include <hip/hip_runtime.h>
#include <math.h>

typedef __attribute__((ext_vector_type(2))) float v2f;
typedef __attribute__((ext_vector_type(8))) float v8f;

#define CHUNK    2048
#define RTHREADS 256
#define NEG_FLT_MAX (-3.402823466e+38f)

// ---------------------------------------------------------------------------
// K1: build S[b,n] = X[b,n,:]·W[0:128], E[b,n] = X[b,n,:]·W[128:256]
// via V_WMMA_F32_16X16X4_F32. One wave32 per 16-row tile, K=128 -> 32 WMMAs.
// A (16x4 f32): lane<16 -> (M=lane, K=0,1); lane>=16 -> (M=lane-16, K=2,3).
// B (4x16 f32): column n = lane&15; col 0 = Ws chunk, col 1 = We chunk, rest 0.
// D (16x16 f32): VGPR v: lanes 0-15 -> M=v, N=lane; lanes 16-31 -> M=8+v.
// EXEC stays all-1s through the WMMA loop (branch-free B construction).
// ---------------------------------------------------------------------------
__global__ __launch_bounds__(32) void se_table_kernel(
    const float* __restrict__ X, const float* __restrict__ W,
    float* __restrict__ S, float* __restrict__ Et, int N)
{
  const int tile      = blockIdx.x;
  const int tilesPerB = N >> 4;
  const int b         = tile / tilesPerB;
  const int row0      = (tile - b * tilesPerB) << 4;
  const int lane      = threadIdx.x & 31;
  const int mrow      = lane & 15;
  const int khalf     = (lane >> 4) << 1;   // 0 or 2
  const int ncol      = lane & 15;          // B/D column this lane carries
  const float selS    = (ncol == 0) ? 1.0f : 0.0f;
  const float selE    = (ncol == 1) ? 1.0f : 0.0f;

  const float* xrow = X + ((size_t)b * N + row0 + mrow) * 128 + khalf;
  const float* w0   = W + khalf;          // Ws chunk base
  const float* w1   = W + 128 + khalf;    // We chunk base

  v8f acc = {};
#pragma unroll
  for (int k = 0; k < 128; k += 4) {
    v2f a  = *(const v2f*)(xrow + k);     // b64 load: (K=khalf, K=khalf+1)
    v2f wS = *(const v2f*)(w0 + k);       // tiny, cache-resident
    v2f wE = *(const v2f*)(w1 + k);
    v2f bb;
    bb.x = wS.x * selS + wE.x * selE;     // branch-free B build (no EXEC games)
    bb.y = wS.y * selS + wE.y * selE;
    acc = __builtin_amdgcn_wmma_f32_16x16x4_f32(
        /*neg_a=*/false, a, /*neg_b=*/false, bb,
        /*c_mod=*/(short)0, acc, /*reuse_a=*/false, /*reuse_b=*/false);
  }

  const int mbase = row0 + ((lane >> 4) << 3);   // +8 for upper lane half
  float* dstS = S  + (size_t)b * N + mbase;
  float* dstE = Et + (size_t)b * N + mbase;
  if (ncol == 0) {
#pragma unroll
    for (int v = 0; v < 8; ++v) dstS[v] = acc[v];
  } else if (ncol == 1) {
#pragma unroll
    for (int v = 0; v < 8; ++v) dstE[v] = acc[v];
  }
}

// ---------------------------------------------------------------------------
// Block tree reductions (fixed order -> deterministic)
// ---------------------------------------------------------------------------
__device__ __forceinline__ float block_reduce_max(float v, float* sm) {
  sm[threadIdx.x] = v; __syncthreads();
  for (int s = RTHREADS >> 1; s > 0; s >>= 1) {
    if ((int)threadIdx.x < s) sm[threadIdx.x] = fmaxf(sm[threadIdx.x], sm[threadIdx.x + s]);
    __syncthreads();
  }
  float r = sm[0]; __syncthreads();
  return r;
}
__device__ __forceinline__ float block_reduce_sum(float v, float* sm) {
  sm[threadIdx.x] = v; __syncthreads();
  for (int s = RTHREADS >> 1; s > 0; s >>= 1) {
    if ((int)threadIdx.x < s) sm[threadIdx.x] = sm[threadIdx.x] + sm[threadIdx.x + s];
    __syncthreads();
  }
  float r = sm[0]; __syncthreads();
  return r;
}

// K2: per-(batch, chunk) partial max of logits (logits recomputed from S/E gathers)
__global__ __launch_bounds__(RTHREADS) void chunk_max_kernel(
    const float* __restrict__ S, const float* __restrict__ Et,
    const int* __restrict__ sidx, const int* __restrict__ eidx,
    int M, int N, float* __restrict__ pmax)
{
  __shared__ float sm[RTHREADS];
  const int b    = blockIdx.y;
  const int m0   = blockIdx.x * CHUNK;
  const int mend = min(m0 + CHUNK, M);
  const float* Sb = S  + (size_t)b * N;
  const float* Eb = Et + (size_t)b * N;
  float v = NEG_FLT_MAX;
  for (int m = m0 + threadIdx.x; m < mend; m += RTHREADS)
    v = fmaxf(v, Sb[sidx[m]] + Eb[eidx[m]]);
  v = block_reduce_max(v, sm);
  if (threadIdx.x == 0) pmax[(size_t)b * gridDim.x + blockIdx.x] = v;
}

// K3: global max (tiny reduction done redundantly per block), exp -> out, partial sums
__global__ __launch_bounds__(RTHREADS) void exp_sum_kernel(
    const float* __restrict__ S, const float* __restrict__ Et,
    const int* __restrict__ sidx, const int* __restrict__ eidx,
    const float* __restrict__ pmax, int M, int N, int NB,
    float* __restrict__ out, float* __restrict__ psum)
{
  __shared__ float sm[RTHREADS];
  const int b = blockIdx.y;
  float gm = NEG_FLT_MAX;
  for (int i = threadIdx.x; i < NB; i += RTHREADS) gm = fmaxf(gm, pmax[(size_t)b * NB + i]);
  gm = block_reduce_max(gm, sm);

  const int m0   = blockIdx.x * CHUNK;
  const int mend = min(m0 + CHUNK, M);
  const float* Sb = S  + (size_t)b * N;
  const float* Eb = Et + (size_t)b * N;
  float* ob = out + (size_t)b * M;
  float lsum = 0.0f;
  for (int m = m0 + threadIdx.x; m < mend; m += RTHREADS) {
    float l = Sb[sidx[m]] + Eb[eidx[m]];
    float e = __expf(l - gm);
    ob[m] = e;
    lsum += e;
  }
  lsum = block_reduce_sum(lsum, sm);
  if (threadIdx.x == 0) psum[(size_t)b * NB + blockIdx.x] = lsum;
}

// K4: global sum (redundant tiny reduction per block), scale out
__global__ __launch_bounds__(RTHREADS) void normalize_kernel(
    const float* __restrict__ psum, int M, int NB, float* __restrict__ out)
{
  __shared__ float sm[RTHREADS];
  const int b = blockIdx.y;
  float gs = 0.0f;
  for (int i = threadIdx.x; i < NB; i += RTHREADS) gs += psum[(size_t)b * NB + i];
  gs = block_reduce_sum(gs, sm);
  const float inv = 1.0f / gs;
  const int m0   = blockIdx.x * CHUNK;
  const int mend = min(m0 + CHUNK, M);
  float* ob = out + (size_t)b * M;
  for (int m = m0 + threadIdx.x; m < mend; m += RTHREADS)
    ob[m] *= inv;
}

// ---------------------------------------------------------------------------
extern "C" void kernel_launch(void* const* d_in, const int* in_sizes, int n_in,
                              void* d_out, int out_size, void* d_ws, size_t ws_size,
                              hipStream_t stream) {
  const float* X    = (const float*)d_in[0];
  const float* W    = (const float*)d_in[1];
  // d_in[2] = bias: shift-invariant under softmax, cancels exactly -> unused
  const int*   sidx = (const int*)d_in[3];
  const int*   eidx = (const int*)d_in[4];
  float*       out  = (float*)d_out;

  const int M = in_sizes[3];              // 100000
  const int E = in_sizes[1] / 2;          // 128 (kernel assumes 128)
  const int B = out_size / M;             // 4
  const int N = in_sizes[0] / (B * E);    // 4096
  const int NB = (M + CHUNK - 1) / CHUNK; // 49

  float* S    = (float*)d_ws;
  float* Et   = S    + (size_t)B * N;
  float* pmax = Et   + (size_t)B * N;
  float* psum = pmax + (size_t)B * NB;

  const int tiles = (B * N) >> 4;         // one wave32 per 16-row tile
  se_table_kernel<<<tiles, 32, 0, stream>>>(X, W, S, Et, N);

  dim3 grid(NB, B);
  chunk_max_kernel<<<grid, RTHREADS, 0, stream>>>(S, Et, sidx, eidx, M, N, pmax);
  exp_sum_kernel  <<<grid, RTHREADS, 0, stream>>>(S, Et, sidx, eidx, pmax, M, N, NB, out, psum);
  normalize_kernel<<<grid, RTHREADS, 0, stream>>>(psum, M, NB, out);
}